// PyramidROIAlign_62706522521663
// MI455X (gfx1250) — compile-verified
//
#include <hip/hip_runtime.h>
#include <math.h>

// PyramidROIAlign for MI455X (gfx1250).
// Bandwidth-bound gather kernel:
//  - CDNA5 async global->LDS copies (global_load_async_to_lds_b128 +
//    s_wait_asynccnt) software-pipeline the 4-corner bilinear gathers,
//    double-buffered in LDS, zero barriers (each thread consumes exactly
//    the data it staged; `p` is uniform per wave32).
//  - Non-temporal output stores keep the ~170MB feature pyramid resident in
//    the 192MB L2, so the 4x corner-row read amplification is served by L2.

#define POOL 7
#define NPIX (POOL * POOL)            // 49 output pixels per ROI
#define CCH 256                       // channels (NHWC innermost)
#define CH4 (CCH / 4)                 // 64 float4 chunks per feature row
#define PIX_PER_STAGE 4               // 4 pixels * 64 chunks = 256 threads
#define NSTAGE ((NPIX + PIX_PER_STAGE - 1) / PIX_PER_STAGE)  // 13

typedef float v4f __attribute__((ext_vector_type(4)));

struct Px {
  int yi0, yi1, xi0, xi1;
  float dy, dx, vmask;
};

__device__ __forceinline__ v4f lerp4(v4f a, v4f b, float t) {
  return a + (b - a) * t;   // componentwise, scalar t broadcast
}

extern "C" __global__ __launch_bounds__(256, 2)
void pyramid_roialign_kernel(const float* __restrict__ rois,
                             const float* __restrict__ p2,
                             const float* __restrict__ p3,
                             const float* __restrict__ p4,
                             const float* __restrict__ p5,
                             const float* __restrict__ meta,
                             float* __restrict__ out,
                             int N, int H2) {
  // double-buffered staging: [buf:2][pixel:4][corner:4][CCH floats] = 32 KB
  __shared__ float sbuf[2 * PIX_PER_STAGE * 4 * CCH];

  const int r = blockIdx.x;          // ROI index 0..B*N-1
  const int b = r / N;               // batch index
  const int t = threadIdx.x;
  const int p = t >> 6;              // pixel slot (uniform per wave32)
  const int k = t & (CH4 - 1);       // float4 chunk within channel row

  // ---- per-ROI scalar setup (block-uniform) ----
  const float y1 = rois[(size_t)r * 4 + 0];
  const float x1 = rois[(size_t)r * 4 + 1];
  const float y2 = rois[(size_t)r * 4 + 2];
  const float x2 = rois[(size_t)r * 4 + 3];

  const float area = meta[b * 11 + 7] * meta[b * 11 + 8];
  const float hw = (y2 - y1) * (x2 - x1);
  const float lf = log2f(sqrtf(fmaxf(hw, 1e-30f)) / (224.0f / sqrtf(area)));
  int lvl = 4 + (int)rintf(lf);      // rintf = round-half-even, matches jnp.round
  lvl = lvl < 2 ? 2 : (lvl > 5 ? 5 : lvl);

  const float* fm = (lvl == 2) ? p2 : (lvl == 3) ? p3 : (lvl == 4) ? p4 : p5;
  const int H = H2 >> (lvl - 2);     // square maps: W == H
  const float fH = (float)(H - 1);
  const size_t bOff = (size_t)b * H * H * CCH;

  auto coords = [&](int P) -> Px {
    const int py = P / POOL, px = P % POOL;
    const float ys = (y1 + ((float)py / (float)(POOL - 1)) * (y2 - y1)) * fH;
    const float xs = (x1 + ((float)px / (float)(POOL - 1)) * (x2 - x1)) * fH;
    const float yf = floorf(ys), xf = floorf(xs);
    Px q;
    q.dy = ys - yf;
    q.dx = xs - xf;
    const int yi = (int)yf, xi = (int)xf;
    q.yi0 = yi < 0 ? 0 : (yi > H - 1 ? H - 1 : yi);
    q.yi1 = (yi + 1) < 0 ? 0 : ((yi + 1) > H - 1 ? H - 1 : yi + 1);
    q.xi0 = xi < 0 ? 0 : (xi > H - 1 ? H - 1 : xi);
    q.xi1 = (xi + 1) < 0 ? 0 : ((xi + 1) > H - 1 ? H - 1 : xi + 1);
    q.vmask = (ys >= 0.0f && ys <= fH && xs >= 0.0f && xs <= fH) ? 1.0f : 0.0f;
    return q;
  };

  // Issue 4 async corner-chunk loads (this thread's own consumption data).
  auto issue = [&](int stage, const Px& q) {
    const unsigned ldsBase = (unsigned)(size_t)(const void*)
        &sbuf[(((stage & 1) * PIX_PER_STAGE + p) * 4) * CCH + k * 4];
    const int yy[2] = {q.yi0, q.yi1};
    const int xx[2] = {q.xi0, q.xi1};
#pragma unroll
    for (int j = 0; j < 4; ++j) {
      const float* g = fm + bOff + ((size_t)yy[j >> 1] * H + xx[j & 1]) * CCH + k * 4;
      const unsigned l = ldsBase + (unsigned)(j * CCH * 4);  // +1KB per corner
      asm volatile("global_load_async_to_lds_b128 %0, %1, off"
                   :: "v"(l), "v"(g) : "memory");
    }
  };

  auto consume = [&](int stage, const Px& q, int P) {
    const int base4 = (((stage & 1) * PIX_PER_STAGE + p) * 4) * CH4 + k;
    const v4f* sb = (const v4f*)sbuf;
    const v4f v00 = sb[base4 + 0 * CH4];
    const v4f v01 = sb[base4 + 1 * CH4];
    const v4f v10 = sb[base4 + 2 * CH4];
    const v4f v11 = sb[base4 + 3 * CH4];
    v4f res = lerp4(lerp4(v00, v01, q.dx), lerp4(v10, v11, q.dx), q.dy);
    res = res * q.vmask;
    // Streaming output: non-temporal so the feature pyramid stays L2-resident.
    v4f* dst = (v4f*)(out + ((size_t)r * NPIX + P) * CCH + k * 4);
    __builtin_nontemporal_store(res, dst);
  };

  // ---- software pipeline over 13 stages of 4 pixels ----
  Px cur = coords(p);                // stage 0: P = p (< 49 always)
  issue(0, cur);
  for (int s = 0; s < NSTAGE; ++s) {
    const int Pn = PIX_PER_STAGE * (s + 1) + p;
    Px nxt = cur;
    if (s + 1 < NSTAGE && Pn < NPIX) {
      nxt = coords(Pn);
      issue(s + 1, nxt);
      // stage s+1 in flight (4 instrs) -> waiting <=4 means stage s landed
      asm volatile("s_wait_asynccnt 4" ::: "memory");
    } else {
      asm volatile("s_wait_asynccnt 0" ::: "memory");
    }
    const int Pc = PIX_PER_STAGE * s + p;
    if (Pc < NPIX) consume(s, cur, Pc);
    cur = nxt;
  }
}

extern "C" void kernel_launch(void* const* d_in, const int* in_sizes, int n_in,
                              void* d_out, int out_size, void* d_ws, size_t ws_size,
                              hipStream_t stream) {
  const float* rois = (const float*)d_in[0];
  const float* p2   = (const float*)d_in[1];
  const float* p3   = (const float*)d_in[2];
  const float* p4   = (const float*)d_in[3];
  const float* p5   = (const float*)d_in[4];
  const float* meta = (const float*)d_in[5];
  float* out = (float*)d_out;

  const int B = in_sizes[5] / 11;
  const int N = in_sizes[0] / (4 * B);
  const long long perMap = (long long)in_sizes[1] / ((long long)B * CCH);
  const int H2 = (int)(sqrt((double)perMap) + 0.5);  // 256 for the reference

  dim3 grid(B * N), block(256);
  pyramid_roialign_kernel<<<grid, block, 0, stream>>>(
      rois, p2, p3, p4, p5, meta, out, N, H2);
}